// ECT_Loss_26044681683496
// MI455X (gfx1250) — compile-verified
//
#include <hip/hip_runtime.h>
#include <hip/hip_bf16.h>

typedef __attribute__((ext_vector_type(2))) float v2f;
typedef __attribute__((ext_vector_type(8))) float v8f;

// ---------------------------------------------------------------------------
// Threefry-2x32 (20 rounds), JAX-compatible (key = (0, 42) for jax.random.key(42))
// ---------------------------------------------------------------------------
__device__ __forceinline__ void threefry2x32(unsigned k0, unsigned k1,
                                             unsigned x0, unsigned x1,
                                             unsigned& o0, unsigned& o1) {
    const unsigned ks0 = k0, ks1 = k1, ks2 = k0 ^ k1 ^ 0x1BD11BDAu;
    const unsigned ks[3] = {ks0, ks1, ks2};
    const int ROT[8] = {13, 15, 26, 6, 17, 29, 16, 24};
    x0 += ks0; x1 += ks1;
#pragma unroll
    for (int g = 0; g < 5; ++g) {
#pragma unroll
        for (int r = 0; r < 4; ++r) {
            int rot = ROT[(g & 1) * 4 + r];
            x0 += x1;
            x1 = (x1 << rot) | (x1 >> (32 - rot));
            x1 ^= x0;
        }
        x0 += ks[(g + 1) % 3];
        x1 += ks[(g + 2) % 3] + (unsigned)(g + 1);
    }
    o0 = x0; o1 = x1;
}

// Giles single-precision erfinv (same approximation family XLA uses for f32 erf_inv)
__device__ __forceinline__ float erfinv_f32(float x) {
    float w = -logf((1.0f - x) * (1.0f + x));
    float p;
    if (w < 5.0f) {
        w = w - 2.5f;
        p = 2.81022636e-08f;
        p = fmaf(p, w, 3.43273939e-07f);
        p = fmaf(p, w, -3.5233877e-06f);
        p = fmaf(p, w, -4.39150654e-06f);
        p = fmaf(p, w, 0.00021858087f);
        p = fmaf(p, w, -0.00125372503f);
        p = fmaf(p, w, -0.00417768164f);
        p = fmaf(p, w, 0.246640727f);
        p = fmaf(p, w, 1.50140941f);
    } else {
        w = sqrtf(w) - 3.0f;
        p = -0.000200214257f;
        p = fmaf(p, w, 0.000100950558f);
        p = fmaf(p, w, 0.00134934322f);
        p = fmaf(p, w, -0.00367342844f);
        p = fmaf(p, w, 0.00573950773f);
        p = fmaf(p, w, -0.0076224613f);
        p = fmaf(p, w, 0.00943887047f);
        p = fmaf(p, w, 1.00167406f);
        p = fmaf(p, w, 2.83297682f);
    }
    return p * x;
}

// V[j] for flat j over a [d, T] gaussian matrix (row-major), JAX normal(key(42))
__global__ void gen_normal_kernel(float* __restrict__ V, int total) {
    int j = blockIdx.x * blockDim.x + threadIdx.x;
    if (j >= total) return;
    unsigned half = (unsigned)(total >> 1);   // total is even for both spaces
    unsigned c0, c1; bool second;
    if ((unsigned)j < half) { c0 = (unsigned)j;        c1 = (unsigned)j + half; second = false; }
    else                    { c0 = (unsigned)j - half; c1 = (unsigned)j;        second = true;  }
    unsigned o0, o1;
    threefry2x32(0u, 42u, c0, c1, o0, o1);
    unsigned bits = second ? o1 : o0;
    // bits -> uniform [0,1): keep 23 mantissa bits, OR exponent of 1.0f, subtract 1
    float f = __uint_as_float((bits >> 9) | 0x3F800000u) - 1.0f;
    const float lo = -0.99999994f;            // nextafter(-1, 0)
    const float hi = 1.0f;
    float u = fmaxf(lo, fmaf(f, hi - lo, lo));
    V[j] = 1.41421356237f * erfinv_f32(u);
}

// Normalize each column (length d) of V:[d, T] to unit L2 norm
__global__ void normalize_cols_kernel(float* __restrict__ V, int d, int T) {
    int t = blockIdx.x * blockDim.x + threadIdx.x;
    if (t >= T) return;
    float ss = 0.0f;
    for (int i = 0; i < d; ++i) {
        float v = V[i * T + t];
        ss = fmaf(v, v, ss);
    }
    float inv = rsqrtf(ss);
    for (int i = 0; i < d; ++i) V[i * T + t] *= inv;
}

// ---------------------------------------------------------------------------
// f32 GEMM via V_WMMA_F32_16X16X4_F32: one wave32 per 16x16 C tile.
//   A: [M, K] row-major,  B: [K, N] row-major,  C: [M, N] row-major.
// A 16x4 layout (ISA 7.12.2): lane%16 = M row; lanes 0-15 hold K={k,k+1},
// lanes 16-31 hold K={k+2,k+3} across the 2 VGPRs. B mirrored with N=lane%16.
// C/D: VGPR g -> row (g + 8*(lane/16)), col = lane%16.
// ---------------------------------------------------------------------------
__global__ void gemm_wmma_f32_kernel(const float* __restrict__ A,
                                     const float* __restrict__ B,
                                     float* __restrict__ C,
                                     int M, int N, int K) {
    int wave  = (int)(blockIdx.x * blockDim.x + threadIdx.x) >> 5;
    int lane  = threadIdx.x & 31;
    int tilesN = N >> 4;
    int tm = (wave / tilesN) << 4;
    int tn = (wave % tilesN) << 4;
    if (tm >= M) return;
    int half = lane >> 4;      // 0 or 1
    int l16  = lane & 15;

    const float* Arow = A + (size_t)(tm + l16) * K;
    v8f acc = {};
    for (int k = 0; k < K; k += 4) {
        int ka = k + half * 2;
        v2f a, b;
        a.x = Arow[ka];
        a.y = Arow[ka + 1];
        b.x = B[(size_t)ka * N + tn + l16];
        b.y = B[(size_t)(ka + 1) * N + tn + l16];
        acc = __builtin_amdgcn_wmma_f32_16x16x4_f32(
            /*neg_a=*/false, a, /*neg_b=*/false, b,
            /*c_mod=*/(short)0, acc, /*reuse_a=*/false, /*reuse_b=*/false);
    }
#pragma unroll
    for (int g = 0; g < 8; ++g) {
        int row = tm + g + half * 8;
        C[(size_t)row * N + tn + l16] = acc[g];
    }
}

// ---------------------------------------------------------------------------
// ECT: E[r, t] = sum_n sigmoid(scale * (lin[r] - NH[n, t]))
// 8 thresholds per thread so each NH element is loaded once per 8 rows.
// ---------------------------------------------------------------------------
__global__ void ect_kernel(const float* __restrict__ NH,  // [Npts, T]
                           float* __restrict__ E,         // [R, T]
                           int Npts, int T, int R) {
    const float radius = 1.0f;
    const float scale  = 500.0f;
    int t  = blockIdx.x * blockDim.x + threadIdx.x;
    int r0 = blockIdx.y * 8;
    if (t >= T) return;
    float step = 2.0f * radius / (float)(R - 1);
    float lin[8];
    float acc[8];
#pragma unroll
    for (int i = 0; i < 8; ++i) {
        lin[i] = fmaf(step, (float)(r0 + i), -radius);
        acc[i] = 0.0f;
    }
    for (int n = 0; n < Npts; ++n) {
        float nh = NH[(size_t)n * T + t];
#pragma unroll
        for (int i = 0; i < 8; ++i) {
            float z = scale * (lin[i] - nh);
            // sigmoid(z) = 1 / (1 + exp(-z))
            acc[i] += __builtin_amdgcn_rcpf(1.0f + __expf(-z));
        }
    }
#pragma unroll
    for (int i = 0; i < 8; ++i) {
        E[(size_t)(r0 + i) * T + t] = acc[i];
    }
}

__global__ void zero_out_kernel(float* __restrict__ p) {
    if (threadIdx.x == 0) p[0] = 0.0f;
}

// mean((A - B)^2): block-level LDS tree reduction, then one fp32 atomic per block
__global__ void mse_kernel(const float* __restrict__ A, const float* __restrict__ B,
                           float* __restrict__ out, int n, float inv_n) {
    __shared__ float sm[256];
    float v = 0.0f;
    for (int i = blockIdx.x * blockDim.x + threadIdx.x; i < n;
         i += gridDim.x * blockDim.x) {
        float d = A[i] - B[i];
        v = fmaf(d, d, v);
    }
    sm[threadIdx.x] = v;
    __syncthreads();
#pragma unroll
    for (int s = 128; s > 0; s >>= 1) {
        if ((int)threadIdx.x < s) sm[threadIdx.x] += sm[threadIdx.x + s];
        __syncthreads();
    }
    if (threadIdx.x == 0) atomicAdd(out, sm[0] * inv_n);
}

// ---------------------------------------------------------------------------
extern "C" void kernel_launch(void* const* d_in, const int* in_sizes, int n_in,
                              void* d_out, int out_size, void* d_ws, size_t ws_size,
                              hipStream_t stream) {
    const float* X1 = (const float*)d_in[0];   // [384, 128] f32
    const float* X2 = (const float*)d_in[1];   // [384, 32]  f32
    const int N  = 384;                        // points = thetas = resolution
    const int D1 = 128;
    const int D2 = 32;

    float* ws  = (float*)d_ws;
    float* V1  = ws;                 // [D1, N]  49152
    float* V2  = V1 + D1 * N;        // [D2, N]  12288
    float* NH1 = V2 + D2 * N;        // [N, N]  147456
    float* NH2 = NH1 + N * N;        // [N, N]
    float* E1  = NH2 + N * N;        // [N, N]
    float* E2  = E1 + N * N;         // [N, N]
    float* out = (float*)d_out;

    // 1) Directions: gaussian then column-normalize
    gen_normal_kernel<<<(D1 * N + 255) / 256, 256, 0, stream>>>(V1, D1 * N);
    gen_normal_kernel<<<(D2 * N + 255) / 256, 256, 0, stream>>>(V2, D2 * N);
    normalize_cols_kernel<<<(N + 127) / 128, 128, 0, stream>>>(V1, D1, N);
    normalize_cols_kernel<<<(N + 127) / 128, 128, 0, stream>>>(V2, D2, N);

    // 2) Projections via WMMA f32: 24x24 = 576 tiles, 8 waves per 256-thread block
    const int tiles  = (N / 16) * (N / 16);       // 576
    const int wavesPerBlock = 256 / 32;           // 8
    const int gblocks = tiles / wavesPerBlock;    // 72
    gemm_wmma_f32_kernel<<<gblocks, 256, 0, stream>>>(X1, V1, NH1, N, N, D1);
    gemm_wmma_f32_kernel<<<gblocks, 256, 0, stream>>>(X2, V2, NH2, N, N, D2);

    // 3) ECT curves (R=N thresholds, 8 per thread in y-grid)
    dim3 eg((N + 127) / 128, N / 8);
    ect_kernel<<<eg, 128, 0, stream>>>(NH1, E1, N, N, N);
    ect_kernel<<<eg, 128, 0, stream>>>(NH2, E2, N, N, N);

    // 4) MSE -> scalar
    zero_out_kernel<<<1, 32, 0, stream>>>(out);
    mse_kernel<<<144, 256, 0, stream>>>(E1, E2, out, N * N, 1.0f / (float)(N * N));
}